// MHA_67765993997200
// MI455X (gfx1250) — compile-verified
//
#include <hip/hip_runtime.h>
#include <hip/hip_bf16.h>
#include <math.h>

// ---------------------------------------------------------------------------
// MHA decode step (Nq=1), B=1024, Nk=512, E=256, h=8, dh=32, fp32.
// HBM-bound: K+V = 1.07 GB @ 23.3 TB/s => ~46 us floor; AI ~ 0.5 FLOP/B.
//   * one workgroup (256 thr / 8 wave32) per batch; K and V each read once.
//   * Phase A: scores via V_WMMA_F32_16X16X4_F32. M=keys, N=heads,
//     contraction over E=256 (64 K-steps of 4). q sits block-diagonally in
//     the B fragment (column n live only when n == e/32) -- implemented as
//     an unconditional ds_load_b64 broadcast * {0,1} lane mask, so no
//     EXEC-predicated loads in the inner loop. A fragments are per-lane
//     global b64 loads straight in WMMA layout (16x reuse per cache line,
//     absorbed by WGP$/L2) -- no LDS staging, no inner-loop barriers.
//     One B fragment feeds 4 key-tiles (4 live v8f accumulators).
//   * Phase B: per-head softmax, one wave per head; unnormalized exp
//     weights kept in LDS, 1/sum per head (normalization folded later).
//   * Phase C: single coalesced fp32 stream of V (128 B per wave per key)
//     with global_prefetch lookahead; prob-mean from the LDS score board.
// ---------------------------------------------------------------------------

typedef float v2f __attribute__((ext_vector_type(2)));
typedef float v8f __attribute__((ext_vector_type(8)));

#define NHEADS 8
#define NKEYS  512
#define EDIM   256

__global__ __launch_bounds__(256) void mha_dec_kernel(
    const float* __restrict__ q,
    const float* __restrict__ k,
    const float* __restrict__ v,
    const int*   __restrict__ mask,
    const int*   __restrict__ clip_p,
    float* __restrict__ out_attn,   // [B, 256]
    float* __restrict__ out_prob)   // [B, 512]
{
    __shared__ float qs[EDIM];            // 1 KB
    __shared__ int   sm[NKEYS];           // 2 KB
    __shared__ float S[NHEADS][NKEYS];    // 16 KB (unnormalized exp weights)
    __shared__ float Linv[NHEADS];

    const int b    = blockIdx.x;
    const int t    = threadIdx.x;
    const int w    = t >> 5;              // wave id (0..7) == head in phase B
    const int lane = t & 31;
    const int g    = lane >> 4;           // WMMA half-wave group (K-pair select)
    const int nl   = lane & 15;           // A: M (key) row / B: N (head) column

    const float clip  = (float)clip_p[0];
    const float scale = 0.17677669529663687f;   // 1/sqrt(32)

    const float* kp = k + (size_t)b * NKEYS * EDIM;
    const float* vp = v + (size_t)b * NKEYS * EDIM;

    // ---- stage q and mask into LDS (coalesced) ----
    qs[t]       = q[(size_t)b * EDIM + t];
    sm[t]       = mask[(size_t)b * NKEYS + t];
    sm[t + 256] = mask[(size_t)b * NKEYS + t + 256];
    __syncthreads();

    // =====================================================================
    // Phase A: S_T[key, head] via fp32 WMMA 16x16x4.
    // Wave w owns keys [64w, 64w+64): 4 tiles of 16 rows, 4 accumulators.
    // A row pointers: rows (64w+nl) + {0,16,32,48}; columns walk E.
    // =====================================================================
    {
        const float* krow = kp + (size_t)(w * 64 + nl) * EDIM;
        v8f acc0 = {}, acc1 = {}, acc2 = {}, acc3 = {};

        for (int hseg = 0; hseg < NHEADS; ++hseg) {       // head segment: e in [32*hseg, 32*hseg+32)
            const float lm = (nl == hseg) ? 1.0f : 0.0f;  // block-diagonal column mask
            #pragma unroll
            for (int j = 0; j < 8; ++j) {                 // 8 K-steps of 4 per segment
                const int ecol = (hseg * 8 + j) * 4 + 2 * g;
                // B fragment: broadcast q pair, masked to column nl==hseg
                const v2f qv = *(const v2f*)(qs + ecol);
                v2f bf; bf.x = qv.x * lm; bf.y = qv.y * lm;
                // A fragments for the 4 key tiles (b64, 16x line reuse)
                const v2f a0 = *(const v2f*)(krow + ecol);
                const v2f a1 = *(const v2f*)(krow + 16 * EDIM + ecol);
                const v2f a2 = *(const v2f*)(krow + 32 * EDIM + ecol);
                const v2f a3 = *(const v2f*)(krow + 48 * EDIM + ecol);
                acc0 = __builtin_amdgcn_wmma_f32_16x16x4_f32(false, a0, false, bf, (short)0, acc0, false, false);
                acc1 = __builtin_amdgcn_wmma_f32_16x16x4_f32(false, a1, false, bf, (short)0, acc1, false, false);
                acc2 = __builtin_amdgcn_wmma_f32_16x16x4_f32(false, a2, false, bf, (short)0, acc2, false, false);
                acc3 = __builtin_amdgcn_wmma_f32_16x16x4_f32(false, a3, false, bf, (short)0, acc3, false, false);
            }
        }

        // Epilogue: D rows = keys (lanes 0-15: M=j, 16-31: M=j+8), col = head.
        const int head = nl;
        if (head < NHEADS) {
            #pragma unroll
            for (int j = 0; j < 8; ++j) {
                const int krel = j + 8 * g;
                v8f* accp[4] = { &acc0, &acc1, &acc2, &acc3 };
                #pragma unroll
                for (int tile = 0; tile < 4; ++tile) {
                    const int key = w * 64 + tile * 16 + krel;
                    float s = (*accp[tile])[j] * scale;
                    if (sm[key] == 1) s = -1.0e9f;
                    s = clip * tanhf(s);
                    S[head][key] = s;
                }
            }
        }
    }
    __syncthreads();

    // =====================================================================
    // Phase B: per-head softmax (wave w -> head w), exp kept unnormalized.
    // =====================================================================
    {
        const int h = w;
        float loc[16];
        float m = -3.4e38f;
        #pragma unroll
        for (int i = 0; i < 16; ++i) {
            loc[i] = S[h][lane + 32 * i];
            m = fmaxf(m, loc[i]);
        }
        #pragma unroll
        for (int o = 16; o > 0; o >>= 1) m = fmaxf(m, __shfl_xor(m, o, 32));
        float sum = 0.0f;
        #pragma unroll
        for (int i = 0; i < 16; ++i) {
            const float e = expf(loc[i] - m);
            sum += e;
            S[h][lane + 32 * i] = e;
        }
        #pragma unroll
        for (int o = 16; o > 0; o >>= 1) sum += __shfl_xor(sum, o, 32);
        if (lane == 0) Linv[h] = 1.0f / sum;
    }
    __syncthreads();

    // =====================================================================
    // Phase C: attn = prob @ V; coalesced single pass over V with prefetch.
    // Thread t owns output column e=t; h is wave-uniform -> LDS broadcast.
    // =====================================================================
    {
        const int e = t;
        const int h = e >> 5;
        const float* vcol = vp + e;
        float a0 = 0.0f;
        for (int k0 = 0; k0 < NKEYS; k0 += 8) {
            const int pf = (k0 + 32 < NKEYS) ? (k0 + 32) : (NKEYS - 1);
            __builtin_prefetch(vcol + (size_t)pf * EDIM, 0, 3);
            #pragma unroll
            for (int kk = k0; kk < k0 + 8; ++kk)
                a0 = fmaf(S[h][kk], vcol[(size_t)kk * EDIM], a0);
        }
        out_attn[(size_t)b * EDIM + e] = a0 * Linv[h];
    }

    // head-mean of probabilities (normalization folded in)
    #pragma unroll
    for (int kk = t; kk < NKEYS; kk += 256) {
        float s = 0.0f;
        #pragma unroll
        for (int hh = 0; hh < NHEADS; ++hh) s += S[hh][kk] * Linv[hh];
        out_prob[(size_t)b * NKEYS + kk] = s * 0.125f;
    }
}

extern "C" void kernel_launch(void* const* d_in, const int* in_sizes, int n_in,
                              void* d_out, int out_size, void* d_ws, size_t ws_size,
                              hipStream_t stream)
{
    const float* q    = (const float*)d_in[0];
    const float* k    = (const float*)d_in[1];
    const float* v    = (const float*)d_in[2];
    const int*   mask = (const int*)d_in[3];
    const int*   clip = (const int*)d_in[4];

    const int B = in_sizes[3] / NKEYS;                    // mask is [B, 512]
    float* out_attn = (float*)d_out;                      // [B, 256]
    float* out_prob = (float*)d_out + (size_t)B * EDIM;   // [B, 512]

    mha_dec_kernel<<<dim3(B), dim3(256), 0, stream>>>(
        q, k, v, mask, clip, out_attn, out_prob);
}